// JKNet_67095979098768
// MI455X (gfx1250) — compile-verified
//
#include <hip/hip_runtime.h>
#include <hip/hip_bf16.h>

// ---------------------------------------------------------------------------
// JKNet (GCN, 4 layers, JK-cat) for gfx1250 / MI455X.
// GEMMs via V_WMMA_F32_16X16X4_F32 (exact fp32, matches reference numerics).
// Edge aggregation via L2-resident gather + global_atomic_add_f32 scatter.
// ---------------------------------------------------------------------------

#define NNODES 50000
#define NEDGES 800000
#define IN_CH  128
#define HIDC   128
#define OUTC   64
#define NLAY   4
#define CATW   (HIDC * NLAY)   // 512

typedef float v2f __attribute__((ext_vector_type(2)));
typedef float v8f __attribute__((ext_vector_type(8)));

// ---------------------------------------------------------------------------
// Generic WMMA GEMM:  C[M, NCOL] = A[M, K] @ W[NCOL, K]^T + bias, optional ReLU.
// Block = one 16-row tile, NCOL/16 waves (one 16x16 tile per wave).
// A tile staged in LDS with +4 float padding (conflict-free ds_load_b64).
// Requires: M % 16 == 0, K % 4 == 0, blockDim.x == 32*(NCOL/16).
// ---------------------------------------------------------------------------
template <int K, int NCOL, bool RELU>
__global__ void gemm_wmma_f32(const float* __restrict__ A, int lda,
                              const float* __restrict__ W,
                              const float* __restrict__ bias,
                              float* __restrict__ C, int ldc)
{
    constexpr int KP = K + 4;                 // padded LDS row stride (floats)
    __shared__ float As[16 * KP];

    const int rowBase = blockIdx.x * 16;
    const int tid     = threadIdx.x;
    const int wave    = tid >> 5;
    const int lane    = tid & 31;
    const int grp     = lane >> 4;            // 0 -> K {0,1}, 1 -> K {2,3}
    const int idx     = lane & 15;            // row (A) / col (B)
    const int col     = wave * 16 + idx;

    // cooperative stage of the 16 x K A-tile into LDS (float4 granularity)
    for (int i = tid * 4; i < 16 * K; i += blockDim.x * 4) {
        const int r = i / K;
        const int c = i % K;
        *(float4*)&As[r * KP + c] =
            *(const float4*)&A[(size_t)(rowBase + r) * lda + c];
    }
    __syncthreads();

    // hint: pull this wave's weight rows toward the caches
    __builtin_prefetch(&W[(size_t)col * K], 0, 3);

    const float cb = bias[col];
    v8f acc;
#pragma unroll
    for (int r = 0; r < 8; ++r) acc[r] = cb;  // D starts as bias broadcast

#pragma unroll 4
    for (int k = 0; k < K; k += 4) {
        // A lane slice: A[idx][k + 2*grp], A[idx][k + 2*grp + 1]
        v2f a = *(const v2f*)&As[idx * KP + k + 2 * grp];
        // B lane slice: W^T[k + 2*grp][col] -> contiguous in W row `col`
        v2f b = *(const v2f*)&W[(size_t)col * K + k + 2 * grp];
        acc = __builtin_amdgcn_wmma_f32_16x16x4_f32(
            /*neg_a=*/false, a, /*neg_b=*/false, b,
            /*c_mod=*/(short)0, acc, /*reuse_a=*/false, /*reuse_b=*/false);
    }

#pragma unroll
    for (int r = 0; r < 8; ++r) {
        const int row = rowBase + r + 8 * grp;
        float v = acc[r];
        if (RELU) v = fmaxf(v, 0.0f);
        C[(size_t)row * ldc + col] = v;
    }
}

// ---------------------------------------------------------------------------
// Degree / edge-norm kernels (deg includes self loop: init 1.0)
// ---------------------------------------------------------------------------
__global__ void deg_init_kernel(float* __restrict__ deg, int n)
{
    int i = blockIdx.x * blockDim.x + threadIdx.x;
    if (i < n) deg[i] = 1.0f;                 // self-loop contribution
}

__global__ void deg_acc_kernel(const int* __restrict__ dst,
                               float* __restrict__ deg, int e)
{
    int i = blockIdx.x * blockDim.x + threadIdx.x;
    if (i < e) atomicAdd(&deg[dst[i]], 1.0f);
}

__global__ void edge_norm_kernel(const int* __restrict__ src,
                                 const int* __restrict__ dst,
                                 const float* __restrict__ deg,
                                 float* __restrict__ nrm, int e)
{
    int i = blockIdx.x * blockDim.x + threadIdx.x;
    if (i >= e) return;
    const float p = deg[src[i]] * deg[dst[i]];
    nrm[i] = (p > 0.0f) ? (1.0f / sqrtf(p)) : 0.0f;   // inf -> 0 guard
}

// ---------------------------------------------------------------------------
// Self-loop init of a concat slice: cat[n, off+c] = (1/deg[n]) * h[n, c]
// one wave per node, 4 channels per lane
// ---------------------------------------------------------------------------
__global__ void self_init_kernel(const float* __restrict__ h,
                                 const float* __restrict__ deg,
                                 float* __restrict__ cat, int off)
{
    int t = blockIdx.x * blockDim.x + threadIdx.x;
    int n = t >> 5;
    if (n >= NNODES) return;
    int lane = t & 31;
    const float ns = 1.0f / deg[n];           // (deg*deg)^-0.5
    float4 hv = *(const float4*)&h[(size_t)n * HIDC + lane * 4];
    float4 ov;
    ov.x = ns * hv.x; ov.y = ns * hv.y; ov.z = ns * hv.z; ov.w = ns * hv.w;
    *(float4*)&cat[(size_t)n * CATW + off + lane * 4] = ov;
}

// ---------------------------------------------------------------------------
// Edge scatter: cat[dst, off+:] += norm[e] * h[src, :]
// one wave per edge, float4 gather per lane, 4 f32 atomics per lane
// ---------------------------------------------------------------------------
__global__ void edge_scatter_kernel(const float* __restrict__ h,
                                    const int* __restrict__ src,
                                    const int* __restrict__ dst,
                                    const float* __restrict__ nrm,
                                    float* __restrict__ cat, int off, int e)
{
    int t = blockIdx.x * blockDim.x + threadIdx.x;
    int ed = t >> 5;
    if (ed >= e) return;
    int lane = t & 31;
    const int   s  = src[ed];
    const int   d  = dst[ed];
    const float nv = nrm[ed];
    float4 hv = *(const float4*)&h[(size_t)s * HIDC + lane * 4];
    float* o = &cat[(size_t)d * CATW + off + lane * 4];
    atomicAdd(o + 0, nv * hv.x);
    atomicAdd(o + 1, nv * hv.y);
    atomicAdd(o + 2, nv * hv.z);
    atomicAdd(o + 3, nv * hv.w);
}

// ---------------------------------------------------------------------------
// In-place ReLU on a concat slice
// ---------------------------------------------------------------------------
__global__ void relu_slice_kernel(float* __restrict__ cat, int off)
{
    int t = blockIdx.x * blockDim.x + threadIdx.x;
    if (t >= NNODES * HIDC) return;
    int n = t / HIDC;
    int c = t - n * HIDC;
    float* p = &cat[(size_t)n * CATW + off + c];
    *p = fmaxf(*p, 0.0f);
}

// ---------------------------------------------------------------------------
// launch
// ---------------------------------------------------------------------------
extern "C" void kernel_launch(void* const* d_in, const int* in_sizes, int n_in,
                              void* d_out, int out_size, void* d_ws, size_t ws_size,
                              hipStream_t stream)
{
    (void)in_sizes; (void)n_in; (void)out_size; (void)ws_size;

    const float* x     = (const float*)d_in[0];   // [N, 128]
    const int*   eidx  = (const int*)  d_in[1];   // [2, E] (int32 under default JAX)
    const float* W_in  = (const float*)d_in[2];   // [128, 128]
    const float* b_in  = (const float*)d_in[3];   // [128]
    const float* Wc    = (const float*)d_in[4];   // [4, 128, 128]
    const float* bc    = (const float*)d_in[5];   // [4, 128]
    const float* W_out = (const float*)d_in[6];   // [64, 512]
    const float* b_out = (const float*)d_in[7];   // [64]
    float*       out   = (float*)d_out;           // [N, 64]

    const int* src = eidx;
    const int* dst = eidx + NEDGES;

    // workspace layout (floats)
    float* ws  = (float*)d_ws;
    float* x0  = ws;                                    // N*128
    float* h   = x0  + (size_t)NNODES * HIDC;           // N*128
    float* cat = h   + (size_t)NNODES * HIDC;           // N*512
    float* deg = cat + (size_t)NNODES * CATW;           // N
    float* nrm = deg + NNODES;                          // E

    const int T = 256;

    // degree with self loops, then per-edge symmetric norm
    deg_init_kernel<<<(NNODES + T - 1) / T, T, 0, stream>>>(deg, NNODES);
    deg_acc_kernel<<<(NEDGES + T - 1) / T, T, 0, stream>>>(dst, deg, NEDGES);
    edge_norm_kernel<<<(NEDGES + T - 1) / T, T, 0, stream>>>(src, dst, deg, nrm, NEDGES);

    // input projection + ReLU:  x0 = relu(x @ W_in^T + b_in)
    gemm_wmma_f32<IN_CH, HIDC, true>
        <<<NNODES / 16, 32 * (HIDC / 16), 0, stream>>>(x, IN_CH, W_in, b_in, x0, HIDC);

    const long waveGridN = ((long)NNODES * 32 + T - 1) / T;
    const long waveGridE = ((long)NEDGES * 32 + T - 1) / T;

    for (int l = 0; l < NLAY; ++l) {
        const float* xin = (l == 0) ? x0 : (cat + (size_t)(l - 1) * HIDC);
        const int    lda = (l == 0) ? HIDC : CATW;
        const int    off = l * HIDC;

        // h = xin @ Wc[l]^T + bc[l]
        gemm_wmma_f32<HIDC, HIDC, false>
            <<<NNODES / 16, 32 * (HIDC / 16), 0, stream>>>(
                xin, lda, Wc + (size_t)l * HIDC * HIDC, bc + (size_t)l * HIDC, h, HIDC);

        // cat[:, off:off+128] = (1/deg) * h   (self loop), then += edge messages
        self_init_kernel<<<waveGridN, T, 0, stream>>>(h, deg, cat, off);
        edge_scatter_kernel<<<waveGridE, T, 0, stream>>>(h, src, dst, nrm, cat, off, NEDGES);

        // ReLU in place -> becomes next layer's input and the JK-cat feature
        relu_slice_kernel<<<((long)NNODES * HIDC + T - 1) / T, T, 0, stream>>>(cat, off);
    }

    // output projection: out = cat @ W_out^T + b_out
    gemm_wmma_f32<CATW, OUTC, false>
        <<<NNODES / 16, 32 * (OUTC / 16), 0, stream>>>(cat, CATW, W_out, b_out, out, OUTC);
}